// SpatialPositionalEncoding2D_3590592660297
// MI455X (gfx1250) — compile-verified
//
#include <hip/hip_runtime.h>
#include <stdint.h>

#define BLOCK 256

// Issue one 16-byte async global->LDS copy per lane (gfx1250 ASYNCcnt path).
__device__ __forceinline__ void async_load_f4(unsigned lds_off, const float4* gptr) {
    asm volatile("global_load_async_to_lds_b128 %0, %1, off"
                 :: "v"(lds_off), "v"((unsigned long long)(uintptr_t)gptr)
                 : "memory");
}

__global__ __launch_bounds__(BLOCK) void pe2d_add_kernel(
        const float* __restrict__ x,
        const int* __restrict__ hp,
        const int* __restrict__ wp,
        float* __restrict__ out,
        int n4) {
    // Double-buffered staging: one private float4 slot per thread per buffer.
    __shared__ float4 stage[2][BLOCK];
    const unsigned tid = threadIdx.x;
    // LDS aperture flat address: low 32 bits are the raw LDS byte offset.
    const unsigned lds0 = (unsigned)(uintptr_t)(&stage[0][tid]);
    const unsigned lds1 = (unsigned)(uintptr_t)(&stage[1][tid]);

    const unsigned W = (unsigned)wp[0];
    const unsigned L = (unsigned)hp[0] * W;   // h*w spatial positions

    const float4* __restrict__ xv = (const float4*)x;
    float4* __restrict__ ov       = (float4*)out;

    const int stride = (int)(gridDim.x * BLOCK);
    int i = (int)(blockIdx.x * BLOCK + tid);
    int buf = 0;

    if (i < n4) async_load_f4(lds0, xv + i);

    for (; i < n4; i += stride) {
        const int nxt = i + stride;
        if (nxt < n4) {
            async_load_f4(buf ? lds0 : lds1, xv + nxt);
            asm volatile("s_wait_asynccnt 0x1" ::: "memory"); // oldest (current) done
        } else {
            asm volatile("s_wait_asynccnt 0x0" ::: "memory"); // drain
        }
        const float4 v = stage[buf][tid];

        // ---- regenerate PE for the 4 consecutive channels of this float4 ----
        const unsigned idx = (unsigned)i;
        const unsigned d0  = (idx & 63u) << 2;   // first channel (4-aligned, d=256)
        const unsigned l   = (idx >> 6) % L;     // spatial position
        unsigned pos, j0;
        if (d0 < 128u) { pos = l / W; j0 = d0; }          // row-encoding half
        else           { pos = l % W; j0 = d0 - 128u; }   // col-encoding half

        // freq(k) = 10000^(-2k/128) = exp2(k * -log2(10000)/64)
        const float C      = -0.20762050593045954f;       // -log2(10000)/64
        const float INV2PI =  0.15915494309189535f;       // HW sin/cos take turns
        const float k0 = (float)(j0 >> 1);                // even: pair base
        const float f0 = __builtin_amdgcn_exp2f(k0 * C);
        const float f1 = __builtin_amdgcn_exp2f(k0 * C + C);
        const float p  = (float)pos;
        const float a0 = p * f0 * INV2PI;
        const float a1 = p * f1 * INV2PI;

        float4 o;
        o.x = v.x + __builtin_amdgcn_sinf(a0);   // channel j0   : sin
        o.y = v.y + __builtin_amdgcn_cosf(a0);   // channel j0+1 : cos
        o.z = v.z + __builtin_amdgcn_sinf(a1);   // channel j0+2 : sin
        o.w = v.w + __builtin_amdgcn_cosf(a1);   // channel j0+3 : cos

        ov[i] = o;                                // global_store_b128
        buf ^= 1;
    }
}

extern "C" void kernel_launch(void* const* d_in, const int* in_sizes, int n_in,
                              void* d_out, int out_size, void* d_ws, size_t ws_size,
                              hipStream_t stream) {
    (void)in_sizes; (void)n_in; (void)d_ws; (void)ws_size;
    const float* x  = (const float*)d_in[0];
    const int*   hp = (const int*)d_in[1];
    const int*   wp = (const int*)d_in[2];
    float*       out = (float*)d_out;

    const int n4 = out_size / 4;                 // float4 elements (d=256 => exact)
    const int iters_per_thread = 8;              // grid-stride pipeline depth
    int blocks = (n4 + BLOCK * iters_per_thread - 1) / (BLOCK * iters_per_thread);
    if (blocks < 1) blocks = 1;

    pe2d_add_kernel<<<blocks, BLOCK, 0, stream>>>(x, hp, wp, out, n4);
}